// DeepSigNet_31610959298966
// MI455X (gfx1250) — compile-verified
//
#include <hip/hip_runtime.h>
#include <hip/hip_bf16.h>

typedef __attribute__((ext_vector_type(2))) float v2f;
typedef __attribute__((ext_vector_type(8))) float v8f;

#define BB      128
#define CIN     64
#define COUT    64
#define LL      512
#define POSTC   384
#define HIDD    1024
#define OUTD    128
#define INDIM   2464   // 64 + 2016 + 1 + 383
#define XCH     448    // CIN + POSTC

// ---------------------------------------------------------------------------
// Kernel 1: Conv1d(64->64,k=3,pad=1) + bias + tanh -> path[b][l][c]
// One WG per (batch, 128-wide L chunk). 8 waves, each owns 4 16x16 out tiles
// (4 independent WMMA accumulator chains per wave).
// GEMM view: A = W[co][kk], kk = ci*3+k (K=192); B = im2col[kk][l].
// ---------------------------------------------------------------------------
__global__ __launch_bounds__(256) void conv_tanh_kernel(
    const float* __restrict__ x, const float* __restrict__ w,
    const float* __restrict__ bias, float* __restrict__ path)
{
    const int b  = blockIdx.y;
    const int l0 = blockIdx.x * 128;

    __shared__ float sX[64][132];  // [ci][l0-1 .. l0+128], 130 used cols

    const float* xb = x + (size_t)b * XCH * LL;
    for (int idx = threadIdx.x; idx < 64 * 130; idx += 256) {
        int ci = idx / 130, li = idx % 130;
        int l  = l0 - 1 + li;
        sX[ci][li] = (l >= 0 && l < LL) ? xb[ci * LL + l] : 0.0f;
    }
    __syncthreads();

    const int wave = threadIdx.x >> 5;
    const int lane = threadIdx.x & 31;
    const int lm   = lane & 15;
    const int half = lane >> 4;
    const int mt     = wave >> 1;        // 0..3  (co tile)
    const int n_base = (wave & 1) * 4;   // n tiles n_base..n_base+3

    const int co_a = mt * 16 + lm;       // A-fragment row (co)

    v8f acc0 = {}, acc1 = {}, acc2 = {}, acc3 = {};

    for (int kk0 = 0; kk0 < 192; kk0 += 4) {
        const int toff = kk0 + half * 2;
        // A frag: contiguous within W row -> float2 load (toff is even)
        v2f a = *(const v2f*)(w + (size_t)co_a * 192 + toff);

        const int ci0 = toff / 3,       k0 = toff - ci0 * 3;
        const int ci1 = (toff + 1) / 3, k1 = (toff + 1) - ci1 * 3;

        {
            int nl = (n_base + 0) * 16 + lm;
            v2f bf; bf.x = sX[ci0][nl + k0]; bf.y = sX[ci1][nl + k1];
            acc0 = __builtin_amdgcn_wmma_f32_16x16x4_f32(false, a, false, bf, (short)0, acc0, false, false);
        }
        {
            int nl = (n_base + 1) * 16 + lm;
            v2f bf; bf.x = sX[ci0][nl + k0]; bf.y = sX[ci1][nl + k1];
            acc1 = __builtin_amdgcn_wmma_f32_16x16x4_f32(false, a, false, bf, (short)0, acc1, false, false);
        }
        {
            int nl = (n_base + 2) * 16 + lm;
            v2f bf; bf.x = sX[ci0][nl + k0]; bf.y = sX[ci1][nl + k1];
            acc2 = __builtin_amdgcn_wmma_f32_16x16x4_f32(false, a, false, bf, (short)0, acc2, false, false);
        }
        {
            int nl = (n_base + 3) * 16 + lm;
            v2f bf; bf.x = sX[ci0][nl + k0]; bf.y = sX[ci1][nl + k1];
            acc3 = __builtin_amdgcn_wmma_f32_16x16x4_f32(false, a, false, bf, (short)0, acc3, false, false);
        }
    }

    // D layout: VGPR r -> M = r + 8*half, N = lm.  Store path[b][l][co].
    float* pb = path + (size_t)b * LL * 64;
    v8f accs[4] = {acc0, acc1, acc2, acc3};
    for (int nt = 0; nt < 4; ++nt) {
        const int l = l0 + (n_base + nt) * 16 + lm;
        #pragma unroll
        for (int r = 0; r < 8; ++r) {
            const int oc = mt * 16 + r + half * 8;
            pb[(size_t)l * 64 + oc] = tanhf(accs[nt][r] + bias[oc]);
        }
    }
}

// ---------------------------------------------------------------------------
// Kernel 2: log-signature (Levy area via 64x64x511 WMMA GEMM) + pool + static
// One WG (16 waves, 512 thr) per batch; each wave owns one 16x16 tile of S,
// accumulated in two interleaved WMMA chains for ILP.
// path streamed through LDS in 4 chunks of up to 129 rows.
// ---------------------------------------------------------------------------
__global__ __launch_bounds__(512) void logsig_feat_kernel(
    const float* __restrict__ x, const float* __restrict__ path,
    float* __restrict__ feat)
{
    const int b = blockIdx.x;
    __shared__ float sP[130][66];
    __shared__ float sS[64][66];
    __shared__ float sRed[512];
    __shared__ float sP0[64];

    const int tid  = threadIdx.x;
    const int wave = tid >> 5;
    const int lane = tid & 31;
    const int lm   = lane & 15;
    const int half = lane >> 4;
    const int it = wave >> 2;   // i tile (0..3)
    const int jt = wave & 3;    // j tile (0..3)

    const float* pb = path + (size_t)b * LL * 64;
    if (tid < 64) sP0[tid] = pb[tid];

    const int ia = it * 16 + lm;   // A row (i)
    const int jb = jt * 16 + lm;   // B col (j)

    v8f accE = {}, accO = {};      // two independent accumulator chains
    for (int chunk = 0; chunk < 4; ++chunk) {
        const int tbase = chunk * 128;
        const int rows  = (chunk == 3) ? 128 : 129;  // rows tbase..tbase+rows-1
        __syncthreads();
        for (int idx = tid; idx < rows * 64; idx += 512) {
            int r = idx >> 6, c = idx & 63;
            sP[r][c] = pb[(size_t)(tbase + r) * 64 + c];
        }
        __syncthreads();

        for (int t0 = 0; t0 < 128; t0 += 8) {
            // ---- even sub-step: t in [t0, t0+4) -------------------------
            {
                const int tl = t0 + half * 2;
                const int tg = tbase + tl;
                const bool v0 = (tg < 511);
                const bool v1 = (tg + 1 < 511);
                v2f a, bf;
                a.x  = v0 ? (sP[tl][ia]     - sP0[ia])    : 0.0f;
                a.y  = v1 ? (sP[tl + 1][ia] - sP0[ia])    : 0.0f;
                bf.x = v0 ? (sP[tl + 1][jb] - sP[tl][jb]) : 0.0f;
                bf.y = v1 ? (sP[tl + 2][jb] - sP[tl + 1][jb]) : 0.0f;
                accE = __builtin_amdgcn_wmma_f32_16x16x4_f32(false, a, false, bf, (short)0, accE, false, false);
            }
            // ---- odd sub-step: t in [t0+4, t0+8) ------------------------
            {
                const int tl = t0 + 4 + half * 2;
                const int tg = tbase + tl;
                const bool v0 = (tg < 511);
                const bool v1 = (tg + 1 < 511);
                v2f a, bf;
                a.x  = v0 ? (sP[tl][ia]     - sP0[ia])    : 0.0f;
                a.y  = v1 ? (sP[tl + 1][ia] - sP0[ia])    : 0.0f;
                bf.x = v0 ? (sP[tl + 1][jb] - sP[tl][jb]) : 0.0f;
                bf.y = v1 ? (sP[tl + 2][jb] - sP[tl + 1][jb]) : 0.0f;
                accO = __builtin_amdgcn_wmma_f32_16x16x4_f32(false, a, false, bf, (short)0, accO, false, false);
            }
        }
    }

    // Deposit S tile for antisymmetrization
    #pragma unroll
    for (int r = 0; r < 8; ++r) {
        const int i = it * 16 + r + half * 8;
        sS[i][jt * 16 + lm] = accE[r] + accO[r];
    }
    __syncthreads();

    float* fb = feat + (size_t)b * INDIM;

    // Level-1 increment
    if (tid < 64) fb[tid] = pb[(size_t)511 * 64 + tid] - sP0[tid];

    // Lyndon pairs (i<j), row-major triu order: 2016 values
    for (int p = tid; p < 2016; p += 512) {
        int i = 0;
        while ((i + 1) * 63 - (i + 1) * i / 2 <= p) ++i;
        const int base = i * 63 - i * (i - 1) / 2;
        const int j = i + 1 + (p - base);
        fb[64 + p] = 0.5f * (sS[i][j] - sS[j][i]);
    }

    // Max-pool of channel CIN over L
    const float* xb = x + (size_t)b * XCH * LL;
    sRed[tid] = xb[(size_t)CIN * LL + tid];
    __syncthreads();
    for (int s = 256; s > 0; s >>= 1) {
        if (tid < s) sRed[tid] = fmaxf(sRed[tid], sRed[tid + s]);
        __syncthreads();
    }
    if (tid == 0) fb[2080] = sRed[0];

    // Static features: channels CIN+1 .. XCH-1 at l=0
    if (tid < POSTC - 1) fb[2081 + tid] = xb[(size_t)(CIN + 1 + tid) * LL];
}

// ---------------------------------------------------------------------------
// Kernels 3-5: fp32 WMMA GEMM  out[m,n] = relu?(A[m,:]·W[n,:] + bias[n])
// One 16x16 tile per wave, dual accumulator chains; 8 waves per WG. M = 128.
// Per-lane global_prefetch of the next 64-wide K chunk of both operand rows.
// ---------------------------------------------------------------------------
template <int N, int K, bool RELU>
__global__ __launch_bounds__(256) void fc_kernel(
    const float* __restrict__ A, const float* __restrict__ W,
    const float* __restrict__ bias, float* __restrict__ out)
{
    const int tid  = threadIdx.x;
    const int wave = tid >> 5;
    const int lane = tid & 31;
    const int lm   = lane & 15;
    const int half = lane >> 4;

    const int gw     = blockIdx.x * 8 + wave;
    const int ntiles = N / 16;
    const int mt = gw / ntiles;
    const int nt = gw % ntiles;

    const float* Ar = A + (size_t)(mt * 16 + lm) * K;   // A-frag row
    const float* Wr = W + (size_t)(nt * 16 + lm) * K;   // B-frag row (W row-major [N][K])

    v8f accE = {}, accO = {};
    for (int k0 = 0; k0 < K; k0 += 8) {
        if ((k0 & 63) == 0 && k0 + 64 < K) {
            // speculative per-lane prefetch of next K chunk (global_prefetch_b8)
            __builtin_prefetch(Ar + k0 + 64, 0, 1);
            __builtin_prefetch(Wr + k0 + 64, 0, 1);
        }
        {
            const int toff = k0 + half * 2;
            v2f a  = *(const v2f*)(Ar + toff);
            v2f bf = *(const v2f*)(Wr + toff);
            accE = __builtin_amdgcn_wmma_f32_16x16x4_f32(false, a, false, bf, (short)0, accE, false, false);
        }
        {
            const int toff = k0 + 4 + half * 2;
            v2f a  = *(const v2f*)(Ar + toff);
            v2f bf = *(const v2f*)(Wr + toff);
            accO = __builtin_amdgcn_wmma_f32_16x16x4_f32(false, a, false, bf, (short)0, accO, false, false);
        }
    }

    #pragma unroll
    for (int r = 0; r < 8; ++r) {
        const int m = mt * 16 + r + half * 8;
        const int n = nt * 16 + lm;
        float v = accE[r] + accO[r] + bias[n];
        if (RELU) v = fmaxf(v, 0.0f);
        out[(size_t)m * N + n] = v;
    }
}

// ---------------------------------------------------------------------------
extern "C" void kernel_launch(void* const* d_in, const int* in_sizes, int n_in,
                              void* d_out, int out_size, void* d_ws, size_t ws_size,
                              hipStream_t stream)
{
    const float* x      = (const float*)d_in[0];
    const float* conv_w = (const float*)d_in[1];
    const float* conv_b = (const float*)d_in[2];
    const float* fc1_w  = (const float*)d_in[3];
    const float* fc1_b  = (const float*)d_in[4];
    const float* fc2_w  = (const float*)d_in[5];
    const float* fc2_b  = (const float*)d_in[6];
    const float* fc3_w  = (const float*)d_in[7];
    const float* fc3_b  = (const float*)d_in[8];

    float* path = (float*)d_ws;                          // B*L*64
    float* feat = path + (size_t)BB * LL * 64;           // B*INDIM
    float* h1   = feat + (size_t)BB * INDIM;             // B*HID
    float* h2   = h1   + (size_t)BB * HIDD;              // B*HID

    conv_tanh_kernel<<<dim3(LL / 128, BB), 256, 0, stream>>>(x, conv_w, conv_b, path);
    logsig_feat_kernel<<<BB, 512, 0, stream>>>(x, path, feat);

    // FC1: 128x1024, K=2464 -> (8*64)/8 = 64 WGs
    fc_kernel<HIDD, INDIM, true><<<64, 256, 0, stream>>>(feat, fc1_w, fc1_b, h1);
    // FC2: 128x1024, K=1024
    fc_kernel<HIDD, HIDD, true><<<64, 256, 0, stream>>>(h1, fc2_w, fc2_b, h2);
    // FC3: 128x128, K=1024 -> (8*8)/8 = 8 WGs
    fc_kernel<OUTD, HIDD, false><<<8, 256, 0, stream>>>(h2, fc3_w, fc3_b, (float*)d_out);
}